// BaseLogicLayer_27075473834525
// MI455X (gfx1250) — compile-verified
//
#include <hip/hip_runtime.h>

#define BATCH   4096
#define IN_DIM  4096
#define OUT_DIM 16384
#define R       4      // batch rows staged in LDS per block (4 * 16KB = 64KB)
#define CT      2048   // output columns per block
#define TPB     256

typedef __attribute__((ext_vector_type(2))) float v2f;
typedef __attribute__((ext_vector_type(4))) float v4f;
typedef __attribute__((ext_vector_type(8))) float v8f;

// B operand for V_WMMA_F32_16X16X4_F32, pre-laid-out per lane.
// BLANE[c][v][lane]: chunk c (K rows 4c..4c+3), VGPR v, lane L.
//   n = L & 15, half = L >> 4, row = 4c + 2*half + v
//   value = (n < 4) ? OP_BASIS[row][n] : 0   (basis padded to 16 columns)
#define Z12 0,0,0,0,0,0,0,0,0,0,0,0
#define PAIR(a0,a1,a2,a3, b0,b1,b2,b3) {a0,a1,a2,a3, Z12, b0,b1,b2,b3, Z12}
__constant__ float BLANE[4][2][32] = {
    { PAIR(0,0,0,0,   0,1,0,-1),  PAIR(0,0,0,1,   0,1,0,0)  },  // rows 0,2 | 1,3
    { PAIR(0,0,1,-1,  0,1,1,-2),  PAIR(0,0,1,0,   0,1,1,-1) },  // rows 4,6 | 5,7
    { PAIR(1,-1,-1,1, 1,0,-1,0),  PAIR(1,-1,-1,2, 1,0,-1,1) },  // rows 8,10| 9,11
    { PAIR(1,-1,0,0,  1,0,0,-1),  PAIR(1,-1,0,1,  1,0,0,0)  },  // rows12,14|13,15
};

// coeffs[j][0..3] = softmax(W[j][0..15]) @ OP_BASIS  via V_WMMA_F32_16X16X4_F32.
// One wave handles 16 gate rows; 4 K-chunks accumulate into one 16x16 f32 D.
__global__ __launch_bounds__(256) void coeffs_kernel(const float* __restrict__ W,
                                                     float* __restrict__ C) {
    const int lane    = threadIdx.x & 31;
    const int wave    = threadIdx.x >> 5;
    const int rowBase = (blockIdx.x * 8 + wave) << 4;   // 16 rows per wave
    const int m       = rowBase + (lane & 15);
    const int half    = lane >> 4;
    const int n       = lane & 15;

    const float4* wr = (const float4*)(W + m * 16);
    float4 q0 = wr[0], q1 = wr[1], q2 = wr[2], q3 = wr[3];
    float w[16] = {q0.x,q0.y,q0.z,q0.w, q1.x,q1.y,q1.z,q1.w,
                   q2.x,q2.y,q2.z,q2.w, q3.x,q3.y,q3.z,q3.w};

    float mx = w[0];
#pragma unroll
    for (int k = 1; k < 16; ++k) mx = fmaxf(mx, w[k]);
    float g[16];
    float s = 0.f;
#pragma unroll
    for (int k = 0; k < 16; ++k) { g[k] = expf(w[k] - mx); s += g[k]; }
    const float inv = 1.0f / s;
#pragma unroll
    for (int k = 0; k < 16; ++k) g[k] *= inv;

    v8f acc = {};
#pragma unroll
    for (int c = 0; c < 4; ++c) {
        // A (16x4): lanes 0-15 -> K {4c,4c+1}; lanes 16-31 -> K {4c+2,4c+3}.
        v2f a;
        a.x = half ? g[4*c + 2] : g[4*c + 0];
        a.y = half ? g[4*c + 3] : g[4*c + 1];
        v2f b;
        b.x = BLANE[c][0][lane];
        b.y = BLANE[c][1][lane];
        acc = __builtin_amdgcn_wmma_f32_16x16x4_f32(
            /*neg_a=*/false, a, /*neg_b=*/false, b,
            /*c_mod=*/(short)0, acc, /*reuse_a=*/false, /*reuse_b=*/false);
    }

    // D: VGPR v, lanes 0-15 -> M=v, N=lane; lanes 16-31 -> M=v+8, N=lane-16.
    if (n < 4) {
#pragma unroll
        for (int v = 0; v < 8; ++v) {
            const int mm = rowBase + v + 8 * half;
            C[mm * 4 + n] = acc[v];
        }
    }
}

// Bandwidth kernel: async-stage R rows of x into LDS (ASYNCcnt path), gather
// a/b from LDS, non-temporal float4 stores so the 256MB output stream does not
// evict the L2-resident x working set.
__global__ __launch_bounds__(TPB) void logic_kernel(const float* __restrict__ x,
                                                    const int* __restrict__ sel,
                                                    const float* __restrict__ C,
                                                    float* __restrict__ out) {
    __shared__ float xs[R * IN_DIM];   // 64 KB
    const int tid  = threadIdx.x;
    const int row0 = blockIdx.y * R;
    const int col0 = blockIdx.x * CT;

    // Stage R contiguous rows of x (R*IN_DIM floats) with async B128 copies.
    {
        const float4* src = (const float4*)(x + (size_t)row0 * IN_DIM);
        const uint32_t lds_base = (uint32_t)(uintptr_t)xs;  // addr[31:0] = LDS offset
        for (int i = tid; i < (R * IN_DIM) / 4; i += TPB) {
            uint64_t gaddr = (uint64_t)(uintptr_t)(src + i);
            uint32_t laddr = lds_base + (uint32_t)i * 16u;
            asm volatile("global_load_async_to_lds_b128 %0, %1, off"
                         :: "v"(laddr), "v"(gaddr) : "memory");
        }
        asm volatile("s_wait_asynccnt 0x0" ::: "memory");
    }
    __syncthreads();

    for (int j = col0 + tid * 4; j < col0 + CT; j += TPB * 4) {
        const int4 s01 = ((const int4*)sel)[(j >> 1) + 0];  // pairs j, j+1
        const int4 s23 = ((const int4*)sel)[(j >> 1) + 1];  // pairs j+2, j+3
        const float4 c0 = ((const float4*)C)[j + 0];
        const float4 c1 = ((const float4*)C)[j + 1];
        const float4 c2 = ((const float4*)C)[j + 2];
        const float4 c3 = ((const float4*)C)[j + 3];
#pragma unroll
        for (int r = 0; r < R; ++r) {
            const float* xr = xs + r * IN_DIM;
            const float a0 = xr[s01.x], b0 = xr[s01.y];
            const float a1 = xr[s01.z], b1 = xr[s01.w];
            const float a2 = xr[s23.x], b2 = xr[s23.y];
            const float a3 = xr[s23.z], b3 = xr[s23.w];
            v4f o;
            o.x = c0.x + c0.y * a0 + c0.z * b0 + c0.w * (a0 * b0);
            o.y = c1.x + c1.y * a1 + c1.z * b1 + c1.w * (a1 * b1);
            o.z = c2.x + c2.y * a2 + c2.z * b2 + c2.w * (a2 * b2);
            o.w = c3.x + c3.y * a3 + c3.z * b3 + c3.w * (a3 * b3);
            __builtin_nontemporal_store(
                o, (v4f*)(out + (size_t)(row0 + r) * OUT_DIM + j));
        }
    }
}

extern "C" void kernel_launch(void* const* d_in, const int* in_sizes, int n_in,
                              void* d_out, int out_size, void* d_ws, size_t ws_size,
                              hipStream_t stream) {
    const float* x   = (const float*)d_in[0];   // (4096, 4096) f32
    const float* W   = (const float*)d_in[1];   // (16384, 16) f32
    const int*   sel = (const int*)d_in[2];     // (16384, 2) i32
    float* out = (float*)d_out;                 // (4096, 16384) f32
    float* C   = (float*)d_ws;                  // 16384*4 f32 = 256 KB scratch

    // 1) per-gate coefficients via WMMA (1024 waves, 8 waves/block)
    coeffs_kernel<<<OUT_DIM / 16 / 8, 256, 0, stream>>>(W, C);

    // 2) bandwidth kernel: grid = (col tiles, row tiles)
    dim3 grid(OUT_DIM / CT, BATCH / R);
    logic_kernel<<<grid, TPB, 0, stream>>>(x, sel, C, out);
}